// NCC_43104291783206
// MI455X (gfx1250) — compile-verified
//
#include <hip/hip_runtime.h>

// ---------------------------------------------------------------------------
// NCC (local, win=9^3) fused tile kernel for MI455X / gfx1250.
//
// - One workgroup per 16^3 output tile; 24^3 haloed I/J tiles staged in LDS
//   (168 KB -- needs CDNA5's 320KB/WGP).
// - Halo staging uses GLOBAL_LOAD_ASYNC_TO_LDS_B32 (ASYNCcnt-tracked, no VGPR
//   stopover); SAME zero-padding handled by DS stores on disjoint addresses.
// - The three separable 9-tap box sums for {I,J,I^2,J^2,IJ} are banded-ones
//   matrix products on V_WMMA_F32_16X16X4_F32 (K=24 -> 6 chained WMMAs).
// - Pointwise NCC formula + reduction in registers; deterministic two-stage
//   reduction (no float atomics).
// Roofline: ~136 MB reads (1.7x halo redundancy) ~= 6 us at 23.3 TB/s;
// ~4.8 G-MAC on the f32 matrix pipe is far below any WMMA ceiling.
// ---------------------------------------------------------------------------

typedef float v2f __attribute__((ext_vector_type(2)));
typedef float v8f __attribute__((ext_vector_type(8)));

namespace {
constexpr int HALO = 4;          // win//2
constexpr int TILE = 16;         // output tile edge
constexpr int EXT  = 24;         // TILE + 2*HALO
constexpr int ND = 2, DD = 160, HH = 192, WW = 160;
constexpr int DT = DD / TILE, HT = HH / TILE, WT = WW / TILE;  // 10,12,10
constexpr int TILES = DT * HT * WT;                            // 1200
constexpr long VOX_PER_N = (long)DD * HH * WW;                 // 4,915,200
constexpr float INV_WINSZ = 1.0f / 729.0f;
constexpr float EPS = 1e-5f;

// LDS layout (float indices). No static LDS in this kernel, so the dynamic
// segment starts at byte offset 0 of the workgroup allocation; byte offsets
// for the async-to-LDS ops are therefore just index*4.
constexpr int SM_I  = 0;
constexpr int SM_J  = SM_I + EXT * EXT * EXT;        // 13824
constexpr int SM_T1 = SM_J + EXT * EXT * EXT;        // 27648  T1[(d*24+h)][w16]
constexpr int SM_T2 = SM_T1 + EXT * EXT * TILE;      // 36864  T2[d24][h16][w16]
constexpr int SM_FLOATS = SM_T2 + EXT * TILE * TILE; // 43008
constexpr size_t SMEM_BYTES = (size_t)SM_FLOATS * sizeof(float); // 172,032 B
}

// Banded ones matrix for the 9-tap box filter on a 24-wide haloed axis:
// out[m] = sum_{t=m}^{m+8} in[t],  m in [0,16), t in [0,24).
__device__ __forceinline__ float bandv(int m, int t) {
  return (t >= m && t <= m + 8) ? 1.0f : 0.0f;
}

__device__ __forceinline__ float quantv(int q, float i, float j) {
  switch (q) {
    case 0:  return i;
    case 1:  return j;
    case 2:  return i * i;
    case 3:  return j * j;
    default: return i * j;
  }
}

// D = A(16x4 f32) * B(4x16 f32) + C(16x16 f32). Native f32 WMMA (CDNA5).
// Operand striping (ISA 7.12.2): lane L supplies A[row=L&15][k] with
// VGPR0/1 = K{0,1} for lanes 0-15, K{2,3} for lanes 16-31; B mirrored with
// L&15 as the column; C/D: VGPR r = row r (lanes 0-15) / r+8 (lanes 16-31).
__device__ __forceinline__ v8f wmma_k4(v2f a, v2f b, v8f c) {
  return __builtin_amdgcn_wmma_f32_16x16x4_f32(
      /*neg_a=*/false, a, /*neg_b=*/false, b,
      /*c_mod=*/(short)0, c, /*reuse_a=*/false, /*reuse_b=*/false);
}

// Async HBM/L2 -> LDS copy, 4 bytes per active lane; tracked by ASYNCcnt.
// VGLOBAL "GV" form: vdst = LDS byte offset, vaddr = 64-bit global address.
__device__ __forceinline__ void async_load_to_lds_b32(unsigned lds_byte_off,
                                                      const void* gaddr) {
  asm volatile("global_load_async_to_lds_b32 %0, %1, off"
               :
               : "v"(lds_byte_off),
                 "v"((unsigned long long)(uintptr_t)gaddr)
               : "memory");
}

__device__ __forceinline__ void wait_async_zero() {
#if __has_builtin(__builtin_amdgcn_s_wait_asynccnt)
  __builtin_amdgcn_s_wait_asynccnt(0);
#else
  asm volatile("s_wait_asynccnt 0" ::: "memory");
#endif
}

__global__ __launch_bounds__(256, 1)
void ncc_tile_kernel(const float* __restrict__ gI,   // y_true
                     const float* __restrict__ gJ,   // y_pred
                     float* __restrict__ partial) {
  extern __shared__ float smem[];
  float* sI = smem + SM_I;
  float* sJ = smem + SM_J;
  float* T1 = smem + SM_T1;
  float* T2 = smem + SM_T2;

  const int wg = blockIdx.x;
  const int n  = wg / TILES;
  int t        = wg % TILES;
  const int dt = t / (HT * WT); t %= (HT * WT);
  const int ht = t / WT;
  const int wt = t % WT;
  const int d0 = dt * TILE - HALO;   // halo-region origin (may be negative)
  const int h0 = ht * TILE - HALO;
  const int w0 = wt * TILE - HALO;

  const int tid = threadIdx.x;
  const long base = (long)n * VOX_PER_N;

  // ---- Haloed tile staging. In-bounds voxels stream straight into LDS via
  // async copies (no VGPR round trip, all 108 requests in flight before the
  // single drain); OOB voxels implement SAME zero-padding via DS stores to
  // disjoint addresses (no ordering hazard with the async traffic).
  // 13824 elements / 256 threads = 54 iterations exactly.
  for (int e = tid; e < EXT * EXT * EXT; e += 256) {
    const int w = e % EXT;
    const int r = e / EXT;
    const int h = r % EXT;
    const int d = r / EXT;
    const int gd = d0 + d, gh = h0 + h, gw = w0 + w;
    const bool in = (gd >= 0 && gd < DD) && (gh >= 0 && gh < HH) &&
                    (gw >= 0 && gw < WW);
    if (in) {
      const long gi = base + ((long)gd * HH + gh) * WW + gw;
      async_load_to_lds_b32((unsigned)((SM_I + e) * 4), gI + gi);
      async_load_to_lds_b32((unsigned)((SM_J + e) * 4), gJ + gi);
    } else {
      sI[e] = 0.0f;
      sJ[e] = 0.0f;
    }
  }
  wait_async_zero();   // each wave drains its own ASYNCcnt...
  __syncthreads();     // ...then the barrier makes every wave's copies visible

  const int lane  = tid & 31;
  const int wave  = tid >> 5;
  const int m     = lane & 15;              // A-row / B-col for this lane
  const int khalf = (lane < 16) ? 0 : 2;    // K sub-slot within a K=4 chunk
  const int rbase = (lane < 16) ? 0 : 8;    // C/D row base for this half-wave

  v8f accR[5][2];   // stage-C results per quantity, 2 column-tiles per wave

#pragma unroll
  for (int q = 0; q < 5; ++q) {
    // ---- Stage A: box sum along W.  Out[w16][dh] = Band x Q[w24][dh].
    // 576 (d,h) columns -> 36 column tiles of 16; waves stride over tiles.
    for (int ct = wave; ct < 36; ct += 8) {
      const int col = ct * 16 + m;          // dh = d*24 + h
      v8f acc = {};
#pragma unroll
      for (int c = 0; c < 6; ++c) {
        const int k0 = 4 * c + khalf;
        v2f a, b;
        a.x = bandv(m, k0);
        a.y = bandv(m, k0 + 1);
        const int i0 = col * EXT + k0;      // sX[(d*24+h)*24 + w_ext]
        b.x = quantv(q, sI[i0], sJ[i0]);
        b.y = quantv(q, sI[i0 + 1], sJ[i0 + 1]);
        acc = wmma_k4(a, b, acc);
      }
      float* dst = &T1[col * TILE];         // T1[dh][w16]
#pragma unroll
      for (int r = 0; r < 8; ++r) dst[rbase + r] = acc[r];
    }
    __syncthreads();

    // ---- Stage B: box sum along H.  Per d: Out[h16][w16] = Band x T1[h24][w16].
    for (int d = wave; d < EXT; d += 8) {
      v8f acc = {};
#pragma unroll
      for (int c = 0; c < 6; ++c) {
        const int k0 = 4 * c + khalf;
        v2f a, b;
        a.x = bandv(m, k0);
        a.y = bandv(m, k0 + 1);
        b.x = T1[(d * EXT + k0) * TILE + m];
        b.y = T1[(d * EXT + k0 + 1) * TILE + m];
        acc = wmma_k4(a, b, acc);
      }
      float* dst = &T2[d * (TILE * TILE) + m];  // T2[d][h][w], col = w = m
#pragma unroll
      for (int r = 0; r < 8; ++r) dst[(rbase + r) * TILE] = acc[r];
    }
    __syncthreads();

    // ---- Stage C: box sum along D.  Out[d16][hw] = Band x T2[d24][hw].
    // 256 (h,w) columns -> 16 tiles; tile index == h. Each wave owns 2.
#pragma unroll
    for (int j = 0; j < 2; ++j) {
      const int ct = wave + j * 8;          // == output h row block
      v8f acc = {};
#pragma unroll
      for (int c = 0; c < 6; ++c) {
        const int k0 = 4 * c + khalf;
        v2f a, b;
        a.x = bandv(m, k0);
        a.y = bandv(m, k0 + 1);
        b.x = T2[k0 * (TILE * TILE) + ct * TILE + m];
        b.y = T2[(k0 + 1) * (TILE * TILE) + ct * TILE + m];
        acc = wmma_k4(a, b, acc);
      }
      accR[q][j] = acc;                     // voxel(d=rbase+r, h=ct, w=m)
    }
    __syncthreads();   // T1/T2 free for next quantity
  }

  // ---- Pointwise NCC + in-register reduction. Each (wave,j,r,lane) maps to
  // the same voxel for all five quantities, so no de-swizzle is needed.
  float lsum = 0.0f;
#pragma unroll
  for (int j = 0; j < 2; ++j) {
#pragma unroll
    for (int r = 0; r < 8; ++r) {
      const float Is = accR[0][j][r];
      const float Js = accR[1][j][r];
      const float I2 = accR[2][j][r];
      const float J2 = accR[3][j][r];
      const float IJ = accR[4][j][r];
      const float cross = IJ - Is * Js * INV_WINSZ;
      const float Iv    = I2 - Is * Is * INV_WINSZ;
      const float Jv    = J2 - Js * Js * INV_WINSZ;
      lsum += 1.0f - (cross * cross) / (Iv * Jv + EPS);
    }
  }
  // wave32 butterfly reduce (fixed order -> deterministic)
#pragma unroll
  for (int off = 16; off > 0; off >>= 1)
    lsum += __shfl_xor(lsum, off, 32);

  if (lane == 0) smem[wave] = lsum;   // LDS free after final barrier above
  __syncthreads();
  if (tid == 0) {
    float s = 0.0f;
#pragma unroll
    for (int w8 = 0; w8 < 8; ++w8) s += smem[w8];
    partial[wg] = s;
  }
}

// Deterministic final reduction: one block per sample, fixed-order tree.
__global__ __launch_bounds__(256)
void ncc_finalize(const float* __restrict__ partial, float* __restrict__ out) {
  __shared__ float red[256];
  const int n = blockIdx.x;
  float s = 0.0f;
  for (int i = threadIdx.x; i < TILES; i += 256) s += partial[n * TILES + i];
  red[threadIdx.x] = s;
  __syncthreads();
  for (int o = 128; o > 0; o >>= 1) {
    if (threadIdx.x < o) red[threadIdx.x] += red[threadIdx.x + o];
    __syncthreads();
  }
  if (threadIdx.x == 0) out[n] = red[0] / (float)VOX_PER_N;
}

extern "C" void kernel_launch(void* const* d_in, const int* in_sizes, int n_in,
                              void* d_out, int out_size, void* d_ws, size_t ws_size,
                              hipStream_t stream) {
  (void)in_sizes; (void)n_in; (void)out_size; (void)ws_size;
  const float* y_pred = (const float*)d_in[0];  // Ji
  const float* y_true = (const float*)d_in[1];  // Ii
  float* out = (float*)d_out;
  float* partial = (float*)d_ws;                // ND*TILES floats, overwritten

  // 172 KB dynamic LDS (> default 64 KB cap); CDNA5 WGP allows up to 320 KB.
  hipFuncSetAttribute((const void*)ncc_tile_kernel,
                      hipFuncAttributeMaxDynamicSharedMemorySize,
                      (int)SMEM_BYTES);

  ncc_tile_kernel<<<ND * TILES, 256, SMEM_BYTES, stream>>>(y_true, y_pred,
                                                           partial);
  ncc_finalize<<<ND, 256, 0, stream>>>(partial, out);
}